// MAMDense_64785286692872
// MI455X (gfx1250) — compile-verified
//
#include <hip/hip_runtime.h>

// Problem constants (reference: IN_FEATURES=512, OUT_FEATURES=256, N=2048)
#define KTOT 512
#define JTOT 256
#define NTOT 2048

// Tiling
#define TN 64            // n-tile per block
#define TJ 64            // j-tile per block
#define KC 32            // k chunk staged in LDS
#define NCHUNK (KTOT / KC)
#define ROWSTR 36        // padded LDS row stride in floats (144 B, 16B-aligned, conflict-free)

// ---------------------------------------------------------------------------
// CDNA5 async global->LDS copy (ASYNCcnt-tracked) with safe fallback.
// ---------------------------------------------------------------------------
#if defined(__gfx1250__) && __has_builtin(__builtin_amdgcn_global_load_async_to_lds_b128)
#define USE_ASYNC 1
#else
#define USE_ASYNC 0
#endif

#if USE_ASYNC
// The builtin expects pointers to 16-byte int vectors (gcc vector_size style),
// AS1 for the global source and AS3 for the LDS destination.
typedef int v4i_gcc __attribute__((vector_size(16)));
typedef __attribute__((address_space(1))) v4i_gcc gv4i_t;
typedef __attribute__((address_space(3))) v4i_gcc lv4i_t;

__device__ __forceinline__ void async_copy16(const float* g, float* l) {
  // Cast through integers (CK-style) to obtain concrete address-space pointers.
  // Generic LDS pointers on gfx1250 carry the LDS offset in addr[31:0].
  gv4i_t* gp = (gv4i_t*)(unsigned long long)g;
  lv4i_t* lp = (lv4i_t*)(unsigned int)(unsigned long long)l;
  __builtin_amdgcn_global_load_async_to_lds_b128(gp, lp, 0, 0);
}

#if __has_builtin(__builtin_amdgcn_s_wait_asynccnt)
#define S_WAIT_ASYNC(n) __builtin_amdgcn_s_wait_asynccnt(n)
#else
#define S_WAIT_ASYNC(n) asm volatile("s_wait_asynccnt %0" ::"i"(n) : "memory")
#endif

#else  // fallback: synchronous copy through VGPRs (global_load_b128 + ds_store_b128)
__device__ __forceinline__ void async_copy16(const float* g, float* l) {
  *(float4*)l = *(const float4*)g;
}
#define S_WAIT_ASYNC(n)
#endif

// ---------------------------------------------------------------------------
// Kernel: C[n,j] = max_k(x[n,k]*w[j,k]) + min_k(x[n,k]*w[j,k]) + bias[j]
// ---------------------------------------------------------------------------
__global__ void __launch_bounds__(256)
mam_dense_kernel(const float* __restrict__ xg, const float* __restrict__ wg,
                 const float* __restrict__ bg, float* __restrict__ outg) {
  __shared__ float xs[2][TN * ROWSTR];   // 2 x 9 KB
  __shared__ float ws[2][TJ * ROWSTR];   // 2 x 9 KB   (36 KB total of 320 KB)

  const int tid = threadIdx.x;
  const int tx  = tid & 15;   // j-lane: handles j = j0 + tx + 16*jj
  const int ty  = tid >> 4;   // n-lane: handles n = n0 + ty + 16*ii
  const int n0  = blockIdx.x * TN;
  const int j0  = blockIdx.y * TJ;

  // Loader mapping: tile = 64 rows x 32 floats = 512 x 16B chunks; 2 per thread.
  const int lrow = tid >> 3;          // 0..31
  const int lcol = (tid & 7) << 2;    // float offset 0,4,...,28 (16B chunks)

  const float* xsrc0 = xg + (size_t)(n0 + lrow)      * KTOT + lcol;
  const float* xsrc1 = xg + (size_t)(n0 + lrow + 32) * KTOT + lcol;
  const float* wsrc0 = wg + (size_t)(j0 + lrow)      * KTOT + lcol;
  const float* wsrc1 = wg + (size_t)(j0 + lrow + 32) * KTOT + lcol;

  auto load_tile = [&](int buf, int k0) {
    async_copy16(xsrc0 + k0, &xs[buf][lrow * ROWSTR + lcol]);
    async_copy16(xsrc1 + k0, &xs[buf][(lrow + 32) * ROWSTR + lcol]);
    async_copy16(wsrc0 + k0, &ws[buf][lrow * ROWSTR + lcol]);
    async_copy16(wsrc1 + k0, &ws[buf][(lrow + 32) * ROWSTR + lcol]);
  };

  float mx[4][4], mn[4][4];
#pragma unroll
  for (int i = 0; i < 4; ++i)
#pragma unroll
    for (int j = 0; j < 4; ++j) {
      mx[i][j] = -__builtin_inff();
      mn[i][j] = __builtin_inff();
    }

  load_tile(0, 0);  // prologue fill

  for (int c = 0; c < NCHUNK; ++c) {
    const int buf = c & 1;
    if (c + 1 < NCHUNK) {
      load_tile(buf ^ 1, (c + 1) * KC);  // prefetch next chunk into other buffer
      S_WAIT_ASYNC(4);                   // chunk c's 4 async ops (in-order) done
    } else {
      S_WAIT_ASYNC(0);
    }
    __syncthreads();  // all waves' chunk-c tile data visible

    const float* __restrict__ xb = xs[buf];
    const float* __restrict__ wb = ws[buf];

#pragma unroll
    for (int kk = 0; kk < KC; kk += 4) {
      float4 X[4], W[4];
#pragma unroll
      for (int i = 0; i < 4; ++i)
        X[i] = *(const float4*)(xb + (ty + 16 * i) * ROWSTR + kk);  // broadcast reads
#pragma unroll
      for (int j = 0; j < 4; ++j)
        W[j] = *(const float4*)(wb + (tx + 16 * j) * ROWSTR + kk);  // bank-conflict-free

#pragma unroll
      for (int i = 0; i < 4; ++i) {
#pragma unroll
        for (int j = 0; j < 4; ++j) {
          const float p0 = X[i].x * W[j].x;
          const float p1 = X[i].y * W[j].y;
          const float p2 = X[i].z * W[j].z;
          const float p3 = X[i].w * W[j].w;
          // short dep chains; clang can fuse into v_max3/min3_num_f32
          const float a = fmaxf(p0, p1), b = fmaxf(p2, p3);
          mx[i][j] = fmaxf(mx[i][j], fmaxf(a, b));
          const float cmin = fminf(p0, p1), d = fminf(p2, p3);
          mn[i][j] = fminf(mn[i][j], fminf(cmin, d));
        }
      }
    }
    __syncthreads();  // protect buf reuse before it is refilled next iteration
  }

  // Epilogue: C = max + min + bias. Stores coalesce across tx (stride-1 in j).
#pragma unroll
  for (int j = 0; j < 4; ++j) {
    const int jj = j0 + tx + 16 * j;
    const float b = bg[jj];
#pragma unroll
    for (int i = 0; i < 4; ++i) {
      const int n = n0 + ty + 16 * i;
      outg[(size_t)n * JTOT + jj] = mx[i][j] + mn[i][j] + b;
    }
  }
}

// ---------------------------------------------------------------------------
// Harness entry point
// ---------------------------------------------------------------------------
extern "C" void kernel_launch(void* const* d_in, const int* in_sizes, int n_in,
                              void* d_out, int out_size, void* d_ws, size_t ws_size,
                              hipStream_t stream) {
  (void)in_sizes; (void)n_in; (void)out_size; (void)d_ws; (void)ws_size;
  const float* x    = (const float*)d_in[0];  // [2048, 512]
  const float* w    = (const float*)d_in[1];  // [256, 512]
  const float* bias = (const float*)d_in[2];  // [256]
  float* out        = (float*)d_out;          // [2048, 256]

  dim3 grid(NTOT / TN, JTOT / TJ);  // (32, 4)
  dim3 block(256);
  mam_dense_kernel<<<grid, block, 0, stream>>>(x, w, bias, out);
}